// SSDLoss_10531259810393
// MI455X (gfx1250) — compile-verified
//
#include <hip/hip_runtime.h>

// Problem constants (from reference setup_inputs)
constexpr int B = 32;
constexpr int N = 32768;
constexpr int C = 21;
constexpr int K = 32;
constexpr int BLK = 256;
constexpr int NBLK = N / BLK;          // 128 blocks per image for per-anchor kernels
constexpr int CE_BLOCKS = (B * N) / 128; // 8192: 8 waves/block * 16 anchors/wave

typedef float v2f __attribute__((ext_vector_type(2)));
typedef float v8f __attribute__((ext_vector_type(8)));

// ---------------------------------------------------------------- kernel 1a
__global__ void k_anchors_max(const float* __restrict__ anchors, float* amax) {
  __shared__ float red[32];
  float m = 0.f;
  for (int i = threadIdx.x; i < N * 4; i += 1024) m = fmaxf(m, fabsf(anchors[i]));
  for (int off = 16; off; off >>= 1) m = fmaxf(m, __shfl_xor(m, off));
  if ((threadIdx.x & 31) == 0) red[threadIdx.x >> 5] = m;
  __syncthreads();
  if (threadIdx.x == 0) {
    float r = red[0];
    for (int w = 1; w < 32; w++) r = fmaxf(r, red[w]);
    *amax = r;
  }
}

// ---------------------------------------------------------------- kernel 1b
// One block per image (128 threads = K*4): gmax, scale, scaled GT, reset winners.
__global__ void k_prep_gt(const float* __restrict__ gt_boxes, const float* __restrict__ amax_p,
                          float* __restrict__ gtScaled, unsigned long long* __restrict__ gtWinner) {
  int b = blockIdx.x;
  __shared__ float red[4];
  const float* g = gt_boxes + b * K * 4;
  float v = fabsf(g[threadIdx.x]);
  for (int off = 16; off; off >>= 1) v = fmaxf(v, __shfl_xor(v, off));
  if ((threadIdx.x & 31) == 0) red[threadIdx.x >> 5] = v;
  __syncthreads();
  float gmax = fmaxf(fmaxf(red[0], red[1]), fmaxf(red[2], red[3]));
  float amax = *amax_p;
  float scale = (gmax > amax * 1.1f) ? (amax / fmaxf(gmax, 1e-12f)) : 1.0f;
  gtScaled[b * K * 4 + threadIdx.x] = g[threadIdx.x] * scale;
  if (threadIdx.x < K) gtWinner[b * K + threadIdx.x] = 0ull;
}

// ---------------------------------------------------------------- kernel 2
// Matching: per-anchor best IoU/gt + per-gt best anchor (packed u64 argmax).
__global__ void k_match(const float* __restrict__ anchors, const float* __restrict__ gtScaled,
                        float* __restrict__ bestIoU, unsigned char* __restrict__ bestGt,
                        unsigned long long* __restrict__ gtWinner) {
  int b = blockIdx.y;
  int a = blockIdx.x * BLK + threadIdx.x;
  __shared__ float4 sg[K];
  __shared__ float sArea[K];
  __shared__ unsigned long long sWin[K];
  if (threadIdx.x < K) {
    const float* p = gtScaled + (b * K + threadIdx.x) * 4;
    float4 bx = {p[0], p[1], p[2], p[3]};
    sg[threadIdx.x] = bx;
    sArea[threadIdx.x] = fmaxf(bx.z - bx.x, 0.f) * fmaxf(bx.w - bx.y, 0.f);
    sWin[threadIdx.x] = 0ull;
  }
  __syncthreads();
  float4 A = ((const float4*)anchors)[a];
  float areaA = fmaxf(A.z - A.x, 0.f) * fmaxf(A.w - A.y, 0.f);
  float best = -1.f;
  int bi = 0;
  for (int k = 0; k < K; k++) {
    float4 g = sg[k];
    float iw = fmaxf(fminf(A.z, g.z) - fmaxf(A.x, g.x), 0.f);
    float ih = fmaxf(fminf(A.w, g.w) - fmaxf(A.y, g.y), 0.f);
    float inter = iw * ih;
    float uni = areaA + sArea[k] - inter;
    float iou = (uni > 0.f) ? (inter / uni) : 0.f;
    if (iou > best) { best = iou; bi = k; }  // first-max tie rule (argmax axis=1)
    // per-gt argmax over anchors: larger iou wins, tie -> smaller anchor idx
    unsigned long long key =
        ((unsigned long long)__float_as_uint(iou) << 32) | (unsigned)(0xFFFFFFFFu - (unsigned)a);
    for (int off = 16; off; off >>= 1) {
      unsigned long long o = __shfl_xor(key, off);
      if (o > key) key = o;
    }
    if ((threadIdx.x & 31) == 0) atomicMax(&sWin[k], key);
  }
  bestIoU[b * N + a] = best;
  bestGt[b * N + a] = (unsigned char)bi;
  __syncthreads();
  if (threadIdx.x < K) atomicMax(&gtWinner[b * K + threadIdx.x], sWin[threadIdx.x]);
}

// ---------------------------------------------------------------- kernel 3a
// Forced assignment, cls targets, smooth-L1 partials, pos-count partials.
__global__ void k_targets(const float* __restrict__ anchors, const float* __restrict__ gtScaled,
                          const int* __restrict__ gt_labels, const float* __restrict__ loc_preds,
                          const float* __restrict__ bestIoU, const unsigned char* __restrict__ bestGt,
                          const unsigned long long* __restrict__ gtWinner,
                          unsigned char* __restrict__ clsT, float* __restrict__ locPart,
                          int* __restrict__ posPart) {
  int b = blockIdx.y;
  int a = blockIdx.x * BLK + threadIdx.x;
  __shared__ unsigned sWinA[K];
  __shared__ float redF[8];
  __shared__ int redI[8];
  if (threadIdx.x < K)
    sWinA[threadIdx.x] = 0xFFFFFFFFu - (unsigned)(gtWinner[b * K + threadIdx.x] & 0xFFFFFFFFull);
  __syncthreads();
  int idx = b * N + a;
  float biou = bestIoU[idx];
  int assigned = bestGt[idx];
  bool pos = biou >= 0.5f;
  for (int k = 0; k < K; k++)
    if (sWinA[k] == (unsigned)a) { assigned = k; pos = true; }  // ascending k: last write wins
  int ct = pos ? (gt_labels[b * K + assigned] + 1) : 0;
  clsT[idx] = (unsigned char)ct;
  float lsum = 0.f;
  if (pos) {
    float4 A = ((const float4*)anchors)[a];
    float acx = (A.x + A.z) * 0.5f, acy = (A.y + A.w) * 0.5f;
    float aw = fmaxf(A.z - A.x, 1e-6f), ah = fmaxf(A.w - A.y, 1e-6f);
    const float* gp = gtScaled + (b * K + assigned) * 4;
    float gcx = (gp[0] + gp[2]) * 0.5f, gcy = (gp[1] + gp[3]) * 0.5f;
    float gw = fmaxf(gp[2] - gp[0], 1e-6f), gh = fmaxf(gp[3] - gp[1], 1e-6f);
    float t[4];
    t[0] = (gcx - acx) / (aw * 0.1f);
    t[1] = (gcy - acy) / (ah * 0.1f);
    t[2] = logf(gw / aw) / 0.2f;
    t[3] = logf(gh / ah) / 0.2f;
    float4 lp = ((const float4*)loc_preds)[idx];
    float l[4] = {lp.x, lp.y, lp.z, lp.w};
    for (int i = 0; i < 4; i++) {
      float d = l[i] - t[i];
      float ad = fabsf(d);
      lsum += (ad < 1.f) ? (0.5f * d * d) : (ad - 0.5f);
    }
  }
  int pc = pos ? 1 : 0;
  for (int off = 16; off; off >>= 1) {
    lsum += __shfl_xor(lsum, off);
    pc += __shfl_xor(pc, off);
  }
  if ((threadIdx.x & 31) == 0) { redF[threadIdx.x >> 5] = lsum; redI[threadIdx.x >> 5] = pc; }
  __syncthreads();
  if (threadIdx.x == 0) {
    float L = 0.f;
    int P = 0;
    for (int i = 0; i < 8; i++) { L += redF[i]; P += redI[i]; }
    int blk = b * NBLK + blockIdx.x;
    locPart[blk] = L;
    posPart[blk] = P;
  }
}

// ---------------------------------------------------------------- kernel 3b
// Cross-entropy via WMMA: sum_k exp(x-mx) computed as exp(X[16x24]) @ ones
// with 6 chained V_WMMA_F32_16X16X4_F32 per 16 anchors per wave.
__global__ void k_ce(const float* __restrict__ cls_preds, const unsigned char* __restrict__ clsT,
                     unsigned* __restrict__ keyOut, float* __restrict__ cePosPart) {
  __shared__ float red[8];
  int lane = threadIdx.x & 31;
  int wave = threadIdx.x >> 5;
  int m = lane & 15, hi = lane >> 4;
  long long a0 = ((long long)blockIdx.x * 8 + wave) * 16;  // global anchor base (b*N+n)
  long long arow = a0 + m;
  const float* row = cls_preds + arow * C;

  // A-matrix layout for f32 16x16x4: lanes 0-15 hold K={0,1}, lanes 16-31 K={2,3};
  // chunk c covers classes 4c..4c+3 -> this lane loads classes 4c+2*hi, +1.
  float vals[12];
  float mx = -__builtin_inff();
#pragma unroll
  for (int c = 0; c < 6; c++) {
    int k0 = 4 * c + 2 * hi;
    float v0 = (k0 < C) ? row[k0] : -__builtin_inff();
    float v1 = (k0 + 1 < C) ? row[k0 + 1] : -__builtin_inff();
    vals[2 * c] = v0;
    vals[2 * c + 1] = v1;
    mx = fmaxf(mx, fmaxf(v0, v1));
  }
  mx = fmaxf(mx, __shfl_xor(mx, 16));  // full 21-class max for anchor m

  v8f acc = {};
  v2f ones;
  ones.x = 1.f;
  ones.y = 1.f;
#pragma unroll
  for (int c = 0; c < 6; c++) {
    v2f av;
    av.x = expf(vals[2 * c] - mx);      // pads: exp(-inf)=0
    av.y = expf(vals[2 * c + 1] - mx);
    acc = __builtin_amdgcn_wmma_f32_16x16x4_f32(false, av, false, ones, (short)0, acc,
                                                false, false);
  }
  // D layout: lo lanes acc[r]=S[r], hi lanes acc[r]=S[8+r]. Want S[lane&15].
  float s = acc[0];
#pragma unroll
  for (int e = 1; e < 8; e++) s = ((lane & 7) == e) ? acc[e] : s;
  float t = __shfl_xor(s, 16);
  if (((lane >> 3) & 1) != hi) s = t;

  int ct = clsT[arow];
  float ce = fmaxf(mx + logf(s) - row[ct], 0.f);  // -logp >= 0

  float cep = 0.f;
  if (hi == 0) {
    // monotone key for negatives (>=1); positives marked 0
    keyOut[arow] = (ct > 0) ? 0u : (__float_as_uint(ce) + 1u);
    if (ct > 0) cep = ce;
  }
  for (int off = 16; off; off >>= 1) cep += __shfl_xor(cep, off);
  if (lane == 0) red[wave] = cep;
  __syncthreads();
  if (threadIdx.x == 0) {
    float S = 0.f;
    for (int i = 0; i < 8; i++) S += red[i];
    cePosPart[blockIdx.x] = S;
  }
}

// ---------------------------------------------------------------- kernel 4
// Per-image radix select of top-num_neg negative CE + deterministic sum.
__global__ void k_select(const unsigned* __restrict__ keys, const int* __restrict__ posPart,
                         float* __restrict__ ceNeg) {
  int b = blockIdx.x;
  const unsigned* kb = keys + (long long)b * N;
  __shared__ unsigned hist[256];
  __shared__ unsigned sPrefix;
  __shared__ int sKK;
  __shared__ int redI[32];
  __shared__ float redF[32];

  int p = 0;
  if (threadIdx.x < NBLK) p = posPart[b * NBLK + threadIdx.x];
  for (int off = 16; off; off >>= 1) p += __shfl_xor(p, off);
  if ((threadIdx.x & 31) == 0) redI[threadIdx.x >> 5] = p;
  __syncthreads();
  if (threadIdx.x == 0) {
    int P = 0;
    for (int i = 0; i < 32; i++) P += redI[i];
    sKK = min(3 * P, N - P);  // num_neg for this image
    sPrefix = 0u;
  }
  __syncthreads();
  if (sKK <= 0) {
    if (threadIdx.x == 0) ceNeg[b] = 0.f;
    return;
  }
  for (int lvl = 3; lvl >= 0; lvl--) {
    for (int i = threadIdx.x; i < 256; i += 1024) hist[i] = 0u;
    __syncthreads();
    unsigned prefix = sPrefix;
    for (int i = threadIdx.x; i < N; i += 1024) {
      unsigned key = kb[i];
      bool match = (lvl == 3) || ((key >> (8 * (lvl + 1))) == prefix);
      if (match) atomicAdd(&hist[(key >> (8 * lvl)) & 255u], 1u);
    }
    __syncthreads();
    if (threadIdx.x == 0) {
      int rem = sKK;
      unsigned c = 0;
      for (int v = 255; v >= 0; v--) {
        if (rem <= (int)(c + hist[v])) {
          sKK = rem - (int)c;                   // rank within equal keys
          sPrefix = (sPrefix << 8) | (unsigned)v;
          break;
        }
        c += hist[v];
      }
    }
    __syncthreads();
  }
  unsigned T = sPrefix;
  int E = sKK;  // number of key==T elements selected (identical ce values)
  float sum = 0.f;
  for (int i = threadIdx.x; i < N; i += 1024) {
    unsigned key = kb[i];
    if (key > T) sum += __uint_as_float(key - 1u);
  }
  for (int off = 16; off; off >>= 1) sum += __shfl_xor(sum, off);
  if ((threadIdx.x & 31) == 0) redF[threadIdx.x >> 5] = sum;
  __syncthreads();
  if (threadIdx.x == 0) {
    float S = 0.f;
    for (int i = 0; i < 32; i++) S += redF[i];
    ceNeg[b] = S + (float)E * __uint_as_float(T - 1u);
  }
}

// ---------------------------------------------------------------- kernel 5
__global__ void k_final(const float* __restrict__ locPart, const int* __restrict__ posPart,
                        const float* __restrict__ cePosPart, const float* __restrict__ ceNeg,
                        float* __restrict__ out) {
  __shared__ float rL[32], rP[32], rN[32];
  __shared__ int rC[32];
  float ls = 0.f, cps = 0.f, cns = 0.f;
  int pc = 0;
  for (int i = threadIdx.x; i < B * NBLK; i += 1024) { ls += locPart[i]; pc += posPart[i]; }
  for (int i = threadIdx.x; i < CE_BLOCKS; i += 1024) cps += cePosPart[i];
  for (int i = threadIdx.x; i < B; i += 1024) cns += ceNeg[i];
  for (int off = 16; off; off >>= 1) {
    ls += __shfl_xor(ls, off);
    cps += __shfl_xor(cps, off);
    cns += __shfl_xor(cns, off);
    pc += __shfl_xor(pc, off);
  }
  int w = threadIdx.x >> 5;
  if ((threadIdx.x & 31) == 0) { rL[w] = ls; rP[w] = cps; rN[w] = cns; rC[w] = pc; }
  __syncthreads();
  if (threadIdx.x == 0) {
    float L = 0.f, CP = 0.f, CN = 0.f;
    int P = 0;
    for (int i = 0; i < 32; i++) { L += rL[i]; CP += rP[i]; CN += rN[i]; P += rC[i]; }
    float np = (float)max(P, 1);
    float loc = L / np;
    float cls = (CP + CN) / np;
    out[0] = loc + cls;
    out[1] = loc;
    out[2] = cls;
  }
}

// ---------------------------------------------------------------- launch
extern "C" void kernel_launch(void* const* d_in, const int* in_sizes, int n_in, void* d_out,
                              int out_size, void* d_ws, size_t ws_size, hipStream_t stream) {
  const float* loc_preds = (const float*)d_in[0];
  const float* cls_preds = (const float*)d_in[1];
  const float* anchors = (const float*)d_in[2];
  const float* gt_boxes = (const float*)d_in[3];
  const int* gt_labels = (const int*)d_in[4];
  float* out = (float*)d_out;

  char* w = (char*)d_ws;
  unsigned* key = (unsigned*)w;                                       // B*N*4      = 4 MiB
  float* bestIoU = (float*)(w + 4194304);                             // B*N*4      = 4 MiB
  unsigned char* bestGt = (unsigned char*)(w + 8388608);              // B*N        = 1 MiB
  unsigned char* clsT = (unsigned char*)(w + 9437184);                // B*N        = 1 MiB
  float* gtScaled = (float*)(w + 10485760);                           // B*K*4*4    = 16 KiB
  unsigned long long* gtWinner = (unsigned long long*)(w + 10502144); // B*K*8      = 8 KiB
  float* amax = (float*)(w + 10510336);                               // 4 B
  float* locPart = (float*)(w + 10510400);                            // 4096*4
  int* posPart = (int*)(w + 10526784);                                // 4096*4
  float* cePosPart = (float*)(w + 10543168);                          // 8192*4
  float* ceNeg = (float*)(w + 10575936);                              // 32*4

  k_anchors_max<<<1, 1024, 0, stream>>>(anchors, amax);
  k_prep_gt<<<B, 128, 0, stream>>>(gt_boxes, amax, gtScaled, gtWinner);
  k_match<<<dim3(NBLK, B), BLK, 0, stream>>>(anchors, gtScaled, bestIoU, bestGt, gtWinner);
  k_targets<<<dim3(NBLK, B), BLK, 0, stream>>>(anchors, gtScaled, gt_labels, loc_preds, bestIoU,
                                               bestGt, gtWinner, clsT, locPart, posPart);
  k_ce<<<CE_BLOCKS, 256, 0, stream>>>(cls_preds, clsT, key, cePosPart);
  k_select<<<B, 1024, 0, stream>>>(key, posPart, ceNeg);
  k_final<<<1, 1024, 0, stream>>>(locPart, posPart, cePosPart, ceNeg, out);
}